// SimpleNetwork_84868553769338
// MI455X (gfx1250) — compile-verified
//
#include <hip/hip_runtime.h>
#include <math.h>

typedef __attribute__((ext_vector_type(2))) float v2f;
typedef __attribute__((ext_vector_type(8))) float v8f;

#define NN 100000
#define NE 3200000

__device__ __forceinline__ float elu1(float x) {
    return x > 0.0f ? x : expm1f(x);
}

// ---------------- degree / normalization ----------------
__global__ void k_init_deg(float* __restrict__ deg) {
    int i = blockIdx.x * blockDim.x + threadIdx.x;
    if (i < NN) deg[i] = 1.0f;                    // self-loop
}

__global__ void k_edge_deg(const int* __restrict__ dst, float* __restrict__ deg) {
    int e = blockIdx.x * blockDim.x + threadIdx.x;
    if (e < NE) atomicAdd(&deg[dst[e]], 1.0f);
}

__global__ void k_dinv(float* __restrict__ deg) {
    int i = blockIdx.x * blockDim.x + threadIdx.x;
    if (i < NN) deg[i] = rsqrtf(deg[i]);          // deg >= 1 always
}

// ---------------- GEMM1: h1 = x[N,8] @ W1[8,64], epilogue scales by dinv and
// seeds the aggregation buffer with the self-loop term ----------------
__global__ __launch_bounds__(128) void k_gemm1(const float* __restrict__ x,
                                               const float* __restrict__ W1,
                                               const float* __restrict__ dinv,
                                               float* __restrict__ h1s,
                                               float* __restrict__ agg1) {
    const int lane = threadIdx.x & 31;
    const int ntile = threadIdx.x >> 5;           // 4 waves -> 4 column tiles of 16
    const int lo = lane & 15, hi = lane >> 4;
    const int m0 = blockIdx.x * 16;               // 100000 = 16 * 6250 exactly

    v8f c = {};
#pragma unroll
    for (int kk = 0; kk < 2; ++kk) {              // K = 8 = 2 x 4
        const int k0 = kk * 4 + hi * 2;
        v2f a, b;
        const float* xr = x + (m0 + lo) * 8 + k0;
        a.x = xr[0];
        a.y = xr[1];
        b.x = W1[k0 * 64 + ntile * 16 + lo];
        b.y = W1[(k0 + 1) * 64 + ntile * 16 + lo];
        c = __builtin_amdgcn_wmma_f32_16x16x4_f32(false, a, false, b,
                                                  (short)0, c, false, false);
    }
#pragma unroll
    for (int v = 0; v < 8; ++v) {
        const int node = m0 + v + hi * 8;
        const int col = ntile * 16 + lo;
        const float val = c[v] * dinv[node];      // pre-scale by dinv[src]
        h1s[node * 64 + col] = val;
        agg1[node * 64 + col] = val;              // self-loop contribution
    }
}

// ---------------- GEMM2: t2 = act1[N,64] @ W2[64,32], same epilogue ----------
__global__ __launch_bounds__(64) void k_gemm2(const float* __restrict__ act1,
                                              const float* __restrict__ W2,
                                              const float* __restrict__ dinv,
                                              float* __restrict__ t2s,
                                              float* __restrict__ agg2) {
    const int lane = threadIdx.x & 31;
    const int ntile = threadIdx.x >> 5;           // 2 waves -> 2 column tiles of 16
    const int lo = lane & 15, hi = lane >> 4;
    const int m0 = blockIdx.x * 16;

    v8f c = {};
#pragma unroll
    for (int kk = 0; kk < 16; ++kk) {             // K = 64 = 16 x 4
        const int k0 = kk * 4 + hi * 2;
        v2f a, b;
        const float* ar = act1 + (m0 + lo) * 64 + k0;
        a.x = ar[0];
        a.y = ar[1];
        b.x = W2[k0 * 32 + ntile * 16 + lo];
        b.y = W2[(k0 + 1) * 32 + ntile * 16 + lo];
        c = __builtin_amdgcn_wmma_f32_16x16x4_f32(false, a, false, b,
                                                  (short)0, c, false, false);
    }
#pragma unroll
    for (int v = 0; v < 8; ++v) {
        const int node = m0 + v + hi * 8;
        const int col = ntile * 16 + lo;
        const float val = c[v] * dinv[node];
        t2s[node * 32 + col] = val;
        agg2[node * 32 + col] = val;
    }
}

// ---------------- edge scatter (gather float4 from src row, 4x f32 atomics) --
__global__ void k_scatter64(const int* __restrict__ src, const int* __restrict__ dst,
                            const float* __restrict__ hs, float* __restrict__ agg) {
    int tid = blockIdx.x * blockDim.x + threadIdx.x;
    if (tid >= NE * 16) return;
    const int e = tid >> 4;
    const int f = (tid & 15) << 2;
    const int s = src[e], d = dst[e];
    const float4 m = *(const float4*)(hs + s * 64 + f);
    float* a = agg + d * 64 + f;
    atomicAdd(a + 0, m.x);
    atomicAdd(a + 1, m.y);
    atomicAdd(a + 2, m.z);
    atomicAdd(a + 3, m.w);
}

__global__ void k_scatter32(const int* __restrict__ src, const int* __restrict__ dst,
                            const float* __restrict__ hs, float* __restrict__ agg) {
    int tid = blockIdx.x * blockDim.x + threadIdx.x;
    if (tid >= NE * 8) return;
    const int e = tid >> 3;
    const int f = (tid & 7) << 2;
    const int s = src[e], d = dst[e];
    const float4 m = *(const float4*)(hs + s * 32 + f);
    float* a = agg + d * 32 + f;
    atomicAdd(a + 0, m.x);
    atomicAdd(a + 1, m.y);
    atomicAdd(a + 2, m.z);
    atomicAdd(a + 3, m.w);
}

// ---------------- post-scale + bias + ELU ----------------
template <int C>
__global__ void k_finalize(const float* __restrict__ agg, const float* __restrict__ dinv,
                           const float* __restrict__ bias, float* __restrict__ out) {
    int i = blockIdx.x * blockDim.x + threadIdx.x;
    if (i >= NN * C) return;
    const int n = i / C;
    const int c = i % C;
    out[i] = elu1(agg[i] * dinv[n] + bias[c]);
}

// ---------------- conv1d(32->16,k3,VALID) + ReLU + FC(16->22) fused ----------
__global__ __launch_bounds__(256) void k_head(const float* __restrict__ h2,
                                              const float* __restrict__ cw,
                                              const float* __restrict__ cb,
                                              const float* __restrict__ fw,
                                              const float* __restrict__ fb,
                                              float* __restrict__ out) {
    __shared__ float s_cw[16 * 32 * 3];
    __shared__ float s_fw[16 * 22];
    __shared__ float s_fb[22];
    __shared__ float s_cb[16];
    for (int i = threadIdx.x; i < 16 * 32 * 3; i += 256) s_cw[i] = cw[i];
    for (int i = threadIdx.x; i < 16 * 22; i += 256) s_fw[i] = fw[i];
    if (threadIdx.x < 22) s_fb[threadIdx.x] = fb[threadIdx.x];
    if (threadIdx.x < 16) s_cb[threadIdx.x] = cb[threadIdx.x];
    __syncthreads();

    const int n = blockIdx.x * blockDim.x + threadIdx.x;
    if (n >= NN - 2) return;

    float acc[16];
#pragma unroll
    for (int o = 0; o < 16; ++o) acc[o] = s_cb[o];

#pragma unroll
    for (int k = 0; k < 3; ++k) {
        const float* row = h2 + (n + k) * 32;
        for (int c = 0; c < 32; ++c) {
            const float v = row[c];
#pragma unroll
            for (int o = 0; o < 16; ++o)
                acc[o] = fmaf(v, s_cw[(o * 32 + c) * 3 + k], acc[o]);
        }
    }
#pragma unroll
    for (int o = 0; o < 16; ++o) acc[o] = fmaxf(acc[o], 0.0f);

    float* op = out + n * 22;
    for (int j = 0; j < 22; ++j) {
        float s = s_fb[j];
#pragma unroll
        for (int o = 0; o < 16; ++o) s = fmaf(acc[o], s_fw[o * 22 + j], s);
        op[j] = s;
    }
}

extern "C" void kernel_launch(void* const* d_in, const int* in_sizes, int n_in,
                              void* d_out, int out_size, void* d_ws, size_t ws_size,
                              hipStream_t stream) {
    (void)in_sizes; (void)n_in; (void)out_size; (void)ws_size;

    const float* x   = (const float*)d_in[0];      // [N, 8]
    const int*   ei  = (const int*)d_in[1];        // [2, E] row-major
    const float* W1  = (const float*)d_in[2];      // [8, 64]
    const float* b1  = (const float*)d_in[3];      // [64]
    const float* W2  = (const float*)d_in[4];      // [64, 32]
    const float* b2  = (const float*)d_in[5];      // [32]
    const float* cw  = (const float*)d_in[6];      // [16, 32, 3]
    const float* cb  = (const float*)d_in[7];      // [16]
    const float* fw  = (const float*)d_in[8];      // [16, 22]
    const float* fb  = (const float*)d_in[9];      // [22]
    float* out = (float*)d_out;                    // [N-2, 22]

    const int* src = ei;
    const int* dst = ei + NE;

    // workspace layout (floats): dinv | buf1[N*64] | agg1[N*64] | buf2[N*32] | agg2[N*32]
    float* ws   = (float*)d_ws;
    float* dinv = ws;
    float* buf1 = dinv + NN;          // h1 pre-scaled, later act1 (ELU output)
    float* agg1 = buf1 + NN * 64;
    float* buf2 = agg1 + NN * 64;     // t2 pre-scaled, later h2 activation
    float* agg2 = buf2 + NN * 32;

    const int B = 256;

    // 1) symmetric normalization coefficients
    k_init_deg<<<(NN + B - 1) / B, B, 0, stream>>>(dinv);
    k_edge_deg<<<(NE + B - 1) / B, B, 0, stream>>>(dst, dinv);
    k_dinv<<<(NN + B - 1) / B, B, 0, stream>>>(dinv);

    // 2) layer 1: WMMA GEMM -> scatter-add -> ELU
    k_gemm1<<<NN / 16, 128, 0, stream>>>(x, W1, dinv, buf1, agg1);
    k_scatter64<<<(NE * 16 + B - 1) / B, B, 0, stream>>>(src, dst, buf1, agg1);
    k_finalize<64><<<(NN * 64 + B - 1) / B, B, 0, stream>>>(agg1, dinv, b1, buf1);

    // 3) layer 2: WMMA GEMM -> scatter-add -> ELU
    k_gemm2<<<NN / 16, 64, 0, stream>>>(buf1, W2, dinv, buf2, agg2);
    k_scatter32<<<(NE * 8 + B - 1) / B, B, 0, stream>>>(src, dst, buf2, agg2);
    k_finalize<32><<<(NN * 32 + B - 1) / B, B, 0, stream>>>(agg2, dinv, b2, buf2);

    // 4) conv1d + relu + fc head
    k_head<<<(NN - 2 + B - 1) / B, B, 0, stream>>>(buf2, cw, cb, fw, fb, out);
}